// PatchesCreate_19318762897712
// MI455X (gfx1250) — compile-verified
//
#include <hip/hip_runtime.h>
#include <stdint.h>

// Patchify: [64, 384, 384, 3] f32 (channels-last) -> [64, 576, 768] f32.
// Pure data movement: 0 FLOPs, ~226 MB traffic -> ~9.7 us floor at 23.3 TB/s.
// CDNA5 async global<->LDS copy path (ASYNCcnt): data never touches VGPRs.
// One patch per wave; patch decompose is forced scalar (SALU), per-lane
// offsets are incremental (no per-chunk divisions).

#define B_    64
#define H_    384
#define W_    384
#define C_    3
#define P_    16
#define G_    (H_ / P_)              // 24 patches per side
#define NPATCH (B_ * G_ * G_)        // 36864
#define PATCH_BYTES 3072             // 16*16*3*4
#define IMG_ROW_BYTES (W_ * C_ * 4)  // 4608

#define THREADS 256
#define WAVES   8                    // one patch per wave
#define CHUNKS  6                    // 6 * 32 lanes * 16B = 3072B = one patch
#define NBLOCKS (NPATCH / WAVES)     // 4608 (exact)

__global__ __launch_bounds__(THREADS)
void patchify_async_kernel(const float* __restrict__ in, float* __restrict__ out) {
    // 8 waves * 3072B = 24576B. Each thread stores from exactly the LDS slot
    // it loaded into -> no cross-wave LDS dependence -> no barrier needed,
    // s_wait_asynccnt alone orders the LDS round-trip.
    __shared__ __align__(16) unsigned char smem[THREADS * CHUNKS * 16];

    const unsigned tid  = threadIdx.x;
    const unsigned lane = tid & 31u;
    // Wave id is uniform across the wave -> force it into an SGPR so the
    // whole patch decompose below compiles to SALU.
    const unsigned wave = (unsigned)__builtin_amdgcn_readfirstlane((int)(tid >> 5));

    // ---- scalar patch decompose (SALU) ----
    const unsigned pg = blockIdx.x * WAVES + wave;   // global patch id
    const unsigned b  = pg / (G_ * G_);
    const unsigned pp = pg - b * (G_ * G_);
    const unsigned gy = pp / G_;
    const unsigned gx = pp - gy * G_;

    // SGPR 64-bit bases for the GVS ("saddr") addressing form.
    const uint64_t in_patch_base = (uint64_t)(uintptr_t)in +
        4ull * ((uint64_t)b * (H_ * W_ * C_) +
                (uint64_t)gy * (P_ * W_ * C_) +
                (uint64_t)gx * (P_ * C_));
    const uint64_t out_patch_base = (uint64_t)(uintptr_t)out + (uint64_t)pg * PATCH_BYTES;

    // Low 32 bits of flat pointer to __shared__ = wave-relative LDS byte addr.
    const unsigned lds_wave = ((unsigned)(uintptr_t)(&smem[0])) + wave * (CHUNKS * 512u);

    // ---- per-lane start within patch: k4 = lane; r0 = lane/12, c4 = lane%12
    // (lane < 32 -> r0 in {0,1,2}; no integer division needed) ----
    const unsigned r0 = (unsigned)(lane >= 12u) + (unsigned)(lane >= 24u);
    unsigned c4      = lane - 12u * r0;                  // float4 within row [0,12)
    unsigned voff_in = r0 * IMG_ROW_BYTES + c4 * 16u;    // bytes into input patch
    unsigned lds_l   = lds_wave + lane * 16u;

    // ---- Phase 1: async gather global -> LDS (strided patch rows) ----
#pragma unroll
    for (int ck = 0; ck < CHUNKS; ++ck) {
        asm volatile("global_load_async_to_lds_b128 %0, %1, %2 th:TH_LOAD_NT"
                     :: "v"(lds_l), "v"(voff_in), "s"(in_patch_base)
                     : "memory");
        // Advance 32 float4 = +2 rows +8 cols; wrap one extra row if c4 >= 4.
        const unsigned wrap = (c4 >= 4u);
        voff_in += wrap ? (3u * IMG_ROW_BYTES - 4u * 16u)   // 13760
                        : (2u * IMG_ROW_BYTES + 8u * 16u);  // 9344
        c4 = c4 + 8u - (wrap ? 12u : 0u);
        lds_l += 512u;
    }

    // Async LDS writes may complete out of order -> drain before reading back.
    asm volatile("s_wait_asynccnt 0" ::: "memory");

    // ---- Phase 2: async scatter LDS -> global (contiguous output) ----
    unsigned voff_out = lane * 16u;
    unsigned lds_s    = lds_wave + lane * 16u;
#pragma unroll
    for (int ck = 0; ck < CHUNKS; ++ck) {
        asm volatile("global_store_async_from_lds_b128 %0, %1, %2 th:TH_STORE_NT"
                     :: "v"(voff_out), "v"(lds_s), "s"(out_patch_base)
                     : "memory");
        voff_out += 512u;
        lds_s    += 512u;
    }

    // Drain stores (S_ENDPGM also implies wait-idle; explicit for clarity).
    asm volatile("s_wait_asynccnt 0" ::: "memory");
}

extern "C" void kernel_launch(void* const* d_in, const int* in_sizes, int n_in,
                              void* d_out, int out_size, void* d_ws, size_t ws_size,
                              hipStream_t stream) {
    (void)in_sizes; (void)n_in; (void)out_size; (void)d_ws; (void)ws_size;
    const float* images = (const float*)d_in[0];
    float* out = (float*)d_out;
    patchify_async_kernel<<<NBLOCKS, THREADS, 0, stream>>>(images, out);
}